// LightGCN_20925080666407
// MI455X (gfx1250) — compile-verified
//
#include <hip/hip_runtime.h>
#include <stdint.h>

#define D 64
#define WAVES_PER_BLOCK 8
#define RING 8   // LDS ring slots per wave (256B each)
#define DIST 7   // prefetch distance (< RING: a slot is never rewritten in the same iter)
#define TILE 256 // scan tile size

// ---------- CSR build ----------
__global__ void k_init_counts(int* cnt_src, int* cnt_dst, int* cursor, int n) {
    int i = blockIdx.x * blockDim.x + threadIdx.x;
    if (i < n) { cnt_src[i] = 0; cnt_dst[i] = 0; cursor[i] = 0; }
}

__global__ void k_hist(const int* __restrict__ src, const int* __restrict__ dst,
                       int* cnt_src, int* cnt_dst, int e) {
    int i = blockIdx.x * blockDim.x + threadIdx.x;
    if (i < e) {
        atomicAdd(&cnt_src[src[i]], 1);
        atomicAdd(&cnt_dst[dst[i]], 1);
    }
}

__global__ void k_norm(const int* __restrict__ cnt_src, float* normed, int n) {
    int i = blockIdx.x * blockDim.x + threadIdx.x;
    if (i < n) {
        int d = cnt_src[i]; if (d < 1) d = 1;
        normed[i] = rsqrtf((float)d);   // clip(deg,1)^-0.5
    }
}

// ---------- Parallel 3-phase exclusive scan of cnt_dst -> row_off ----------
// Phase 1: per-tile sums (fully parallel)
__global__ __launch_bounds__(TILE) void k_tile_sums(const int* __restrict__ cnt,
                                                    int* __restrict__ tsum, int n) {
    __shared__ int sh[TILE];
    int t = threadIdx.x;
    int i = blockIdx.x * TILE + t;
    sh[t] = (i < n) ? cnt[i] : 0;
    __syncthreads();
    for (int off = TILE / 2; off > 0; off >>= 1) {
        if (t < off) sh[t] += sh[t + off];
        __syncthreads();
    }
    if (t == 0) tsum[blockIdx.x] = sh[0];
}

// Phase 2: one block scans up to 1024 tile sums (586 for this problem)
__global__ __launch_bounds__(1024) void k_scan_tiles(const int* __restrict__ tsum,
                                                     int* __restrict__ toff,
                                                     int* __restrict__ row_off,
                                                     int nt, int n) {
    __shared__ int sh[1024];
    int t = threadIdx.x;
    int v = (t < nt) ? tsum[t] : 0;
    sh[t] = v;
    __syncthreads();
    for (int off = 1; off < 1024; off <<= 1) {
        int x = (t >= off) ? sh[t - off] : 0;
        __syncthreads();
        sh[t] += x;
        __syncthreads();
    }
    if (t < nt) toff[t] = sh[t] - v;        // exclusive tile offsets
    if (t == 1023) row_off[n] = sh[1023];   // grand total = E
}

// Phase 3: per-tile local exclusive scan + tile offset (fully parallel)
__global__ __launch_bounds__(TILE) void k_scan_local(const int* __restrict__ cnt,
                                                     const int* __restrict__ toff,
                                                     int* __restrict__ row_off, int n) {
    __shared__ int sh[TILE];
    int t = threadIdx.x;
    int i = blockIdx.x * TILE + t;
    int v = (i < n) ? cnt[i] : 0;
    sh[t] = v;
    __syncthreads();
    for (int off = 1; off < TILE; off <<= 1) {
        int x = (t >= off) ? sh[t - off] : 0;
        __syncthreads();
        sh[t] += x;
        __syncthreads();
    }
    if (i < n) row_off[i] = toff[blockIdx.x] + sh[t] - v;
}

__global__ void k_fill(const int* __restrict__ src, const int* __restrict__ dst,
                       const int* __restrict__ row_off, int* cursor, int* csr, int e) {
    int i = blockIdx.x * blockDim.x + threadIdx.x;
    if (i < e) {
        int d = dst[i];
        int p = atomicAdd(&cursor[d], 1);
        csr[row_off[d] + p] = src[i];
    }
}

// emb0 = concat(users, items); acc(d_out) = 0.25 * emb0 (mean over K+1 layers folded in)
__global__ void k_init_emb(const float* __restrict__ users, const float* __restrict__ items,
                           float* embA, float* acc, int nu_elems, int total) {
    int i = blockIdx.x * blockDim.x + threadIdx.x;
    if (i < total) {
        float v = (i < nu_elems) ? users[i] : items[i - nu_elems];
        embA[i] = v;
        acc[i]  = 0.25f * v;
    }
}

// ---------- One propagation layer: wave-per-node gather-reduce, async LDS pipeline ----------
__global__ __launch_bounds__(256) void k_layer(
    const float* __restrict__ embIn, float* __restrict__ embOut,
    const int* __restrict__ row_off, const int* __restrict__ csr,
    const float* __restrict__ normed, float* __restrict__ acc, int n)
{
    __shared__ float ring[WAVES_PER_BLOCK][RING][D];   // 16 KB/block
    const int wv   = threadIdx.x >> 5;
    const int lane = threadIdx.x & 31;
    const int node = blockIdx.x * WAVES_PER_BLOCK + wv;
    if (node >= n) return;

    const int s0 = row_off[node];
    const int s1 = row_off[node + 1];
    const int len = s1 - s0;

    float a0 = 0.f, a1 = 0.f;
    const uint32_t lds_base =
        (uint32_t)(uintptr_t)(&ring[wv][0][0]) + (uint32_t)(lane * 8);

    // prefetch up to DIST rows: 32 lanes x b64 = one 256B embedding row per edge
    const int pre = (len < DIST) ? len : DIST;
    for (int j = 0; j < pre; ++j) {
        int s = csr[s0 + j];
        uint64_t ga = (uint64_t)(uintptr_t)(embIn + (size_t)s * D + lane * 2);
        uint32_t la = lds_base + (uint32_t)((j & (RING - 1)) * (D * 4));
        asm volatile("global_load_async_to_lds_b64 %0, %1, off"
                     :: "v"(la), "v"(ga) : "memory");
    }

    int j = 0;
    // steady state: wait for oldest, consume, refill a different ring slot
    for (; j + DIST < len; ++j) {
        asm volatile("s_wait_asynccnt 6" ::: "memory");   // oldest load complete
        int s = csr[s0 + j];
        float w = normed[s];
        const float* slot = &ring[wv][j & (RING - 1)][lane * 2];
        a0 += w * slot[0];
        a1 += w * slot[1];

        int jn = j + DIST;
        int s2 = csr[s0 + jn];
        uint64_t ga = (uint64_t)(uintptr_t)(embIn + (size_t)s2 * D + lane * 2);
        uint32_t la = lds_base + (uint32_t)((jn & (RING - 1)) * (D * 4));
        asm volatile("global_load_async_to_lds_b64 %0, %1, off"
                     :: "v"(la), "v"(ga) : "memory");
    }
    // drain
    asm volatile("s_wait_asynccnt 0" ::: "memory");
    for (; j < len; ++j) {
        int s = csr[s0 + j];
        float w = normed[s];
        const float* slot = &ring[wv][j & (RING - 1)][lane * 2];
        a0 += w * slot[0];
        a1 += w * slot[1];
    }

    const float wn = normed[node];
    const float r0 = a0 * wn, r1 = a1 * wn;
    const size_t idx = (size_t)node * D + lane * 2;

    *(float2*)(embOut + idx) = make_float2(r0, r1);   // next layer input
    float2 o = *(const float2*)(acc + idx);           // wave owns this row: safe RMW
    o.x += 0.25f * r0;
    o.y += 0.25f * r1;
    *(float2*)(acc + idx) = o;
}

// ---------- launch ----------
extern "C" void kernel_launch(void* const* d_in, const int* in_sizes, int n_in,
                              void* d_out, int out_size, void* d_ws, size_t ws_size,
                              hipStream_t stream) {
    const float* users = (const float*)d_in[0];
    const float* items = (const float*)d_in[1];
    const int*   src   = (const int*)d_in[2];
    const int*   dst   = (const int*)d_in[3];

    const int nu = in_sizes[0] / D;
    const int ni = in_sizes[1] / D;
    const int N  = nu + ni;
    const int E  = in_sizes[2];
    const int NT = (N + TILE - 1) / TILE;   // number of scan tiles (586)

    char* p = (char*)d_ws;
    auto alloc = [&](size_t bytes) -> void* {
        void* r = (void*)p;
        p += (bytes + 255) & ~(size_t)255;
        return r;
    };
    float* normed  = (float*)alloc((size_t)N * 4);
    int*   cnt_src = (int*)  alloc((size_t)N * 4);
    int*   cnt_dst = (int*)  alloc((size_t)N * 4);
    int*   row_off = (int*)  alloc((size_t)(N + 1) * 4);
    int*   cursor  = (int*)  alloc((size_t)N * 4);
    int*   tsum    = (int*)  alloc((size_t)NT * 4);
    int*   toff    = (int*)  alloc((size_t)NT * 4);
    int*   csr     = (int*)  alloc((size_t)E * 4);
    float* embA    = (float*)alloc((size_t)N * D * 4);
    float* embB    = (float*)alloc((size_t)N * D * 4);
    float* acc     = (float*)d_out;

    const int tb = 256;
    k_init_counts<<<(N + tb - 1) / tb, tb, 0, stream>>>(cnt_src, cnt_dst, cursor, N);
    k_hist<<<(E + tb - 1) / tb, tb, 0, stream>>>(src, dst, cnt_src, cnt_dst, E);
    k_norm<<<(N + tb - 1) / tb, tb, 0, stream>>>(cnt_src, normed, N);

    k_tile_sums<<<NT, TILE, 0, stream>>>(cnt_dst, tsum, N);
    k_scan_tiles<<<1, 1024, 0, stream>>>(tsum, toff, row_off, NT, N);
    k_scan_local<<<NT, TILE, 0, stream>>>(cnt_dst, toff, row_off, N);

    k_fill<<<(E + tb - 1) / tb, tb, 0, stream>>>(src, dst, row_off, cursor, csr, E);

    const int total = N * D;
    k_init_emb<<<(total + tb - 1) / tb, tb, 0, stream>>>(users, items, embA, acc, nu * D, total);

    const int blocks = (N + WAVES_PER_BLOCK - 1) / WAVES_PER_BLOCK;
    k_layer<<<blocks, 256, 0, stream>>>(embA, embB, row_off, csr, normed, acc, N);
    k_layer<<<blocks, 256, 0, stream>>>(embB, embA, row_off, csr, normed, acc, N);
    k_layer<<<blocks, 256, 0, stream>>>(embA, embB, row_off, csr, normed, acc, N);
}